// SpectralConv2D_16527034155165
// MI455X (gfx1250) — compile-verified
//
#include <hip/hip_runtime.h>

// SpectralConv2D (FNO) for MI455X / gfx1250.
// Strategy: replace rfft2/irfft2 with partial-DFT GEMMs on the f32 WMMA pipe.
// B=16, Cin=Cout=32, H=W=256, modes m1=m2=16.
//
// HBM traffic: read x once (134 MB, non-temporal) + write out once (134 MB,
// non-temporal). All intermediates (Y 16.8MB, Z/T 4MB, U 16.8MB, tables 192KB)
// stay L2-resident (192 MB) with default RT temporal hints.

typedef __attribute__((ext_vector_type(2))) float v2f;
typedef __attribute__((ext_vector_type(8))) float v8f;

__device__ __forceinline__ v8f wmma4(v2f a, v2f b, v8f c) {
  // V_WMMA_F32_16X16X4_F32 : D = A(16x4) * B(4x16) + C(16x16)
  return __builtin_amdgcn_wmma_f32_16x16x4_f32(false, a, false, b, (short)0, c,
                                               false, false);
}

__device__ __forceinline__ v2f ld2(const float* p) { return *(const v2f*)p; }
// Non-temporal (TH=NT) load: for stream-once data that must not evict the
// L2-resident intermediates.
__device__ __forceinline__ v2f ld2_nt(const float* p) {
  return __builtin_nontemporal_load((const v2f*)p);
}

// ---------------------------------------------------------------------------
// K0: build twiddle tables in workspace (float layout, offsets in floats):
//   FT  [32 n][256 w] : n<16: cos(2pi n w/256)        ; n>=16: -sin (forward W)
//   E2c [32 r][256 h] : cos(2pi ky(r) h/256), ky = r<16 ? r : 224+r
//   E2s [32 r][256 h] : sin(...)
//   E4c [256 h][32 r] : cos(2pi ky(r) h/256)   (inverse H)
//   E4s [256 h][32 r] : sin(...)
//   GT  [256 w][32 n] : n<16:  a_k*cos(2pi k w/256)/65536
//                       n>=16: -a_k*sin(2pi k w/256)/65536, a_0=1 else 2
// ---------------------------------------------------------------------------
__global__ __launch_bounds__(256) void k0_tables(float* __restrict__ tab) {
  int idx = blockIdx.x * 256 + threadIdx.x;
  if (idx >= 49152) return;
  const float step = 6.28318530717958647692f / 256.0f;
  float v;
  if (idx < 8192) {  // FT
    int n = idx >> 8, w = idx & 255;
    float th = step * (float)(((n & 15) * w) & 255);
    v = (n < 16) ? cosf(th) : -sinf(th);
  } else if (idx < 24576) {  // E2c / E2s
    int j = idx - 8192;
    int is_s = j >= 8192;
    j &= 8191;
    int r = j >> 8, h = j & 255;
    int ky = (r < 16) ? r : (224 + r);
    float th = step * (float)((ky * h) & 255);
    v = is_s ? sinf(th) : cosf(th);
  } else if (idx < 40960) {  // E4c / E4s
    int j = idx - 24576;
    int is_s = j >= 8192;
    j &= 8191;
    int h = j >> 5, r = j & 31;
    int ky = (r < 16) ? r : (224 + r);
    float th = step * (float)((ky * h) & 255);
    v = is_s ? sinf(th) : cosf(th);
  } else {  // GT
    int j = idx - 40960;
    int w = j >> 5, n = j & 31;
    int kx = n & 15;
    float th = step * (float)((kx * w) & 255);
    float sc = ((kx == 0) ? 1.0f : 2.0f) * (1.0f / 65536.0f);  // 1/(H*W), alpha
    v = (n < 16) ? sc * cosf(th) : -sc * sinf(th);
  }
  tab[idx] = v;
}

// ---------------------------------------------------------------------------
// K1: forward partial DFT over W.
// Y[row=(b,i,h)][n] (n: 0..15 = Re, 16..31 = Im), GEMM [131072,256]x[256,32].
// One wave per 16-row tile; K stepped by 4; 2 N-tiles.
// x is streamed with NT hints (read once); FT hits L1/L2; Y stays RT.
// ---------------------------------------------------------------------------
__global__ __launch_bounds__(256) void k1_dft_w(const float* __restrict__ x,
                                                const float* __restrict__ FT,
                                                float* __restrict__ Y) {
  const int wave = blockIdx.x * 8 + (threadIdx.x >> 5);  // 0..8191
  const int lane = threadIdx.x & 31;
  const int half = lane >> 4, l16 = lane & 15;
  const long rowbase = (long)wave * 16;
  const float* arow = x + (rowbase + l16) * 256 + 2 * half;
  const float* b0 = FT + l16 * 256 + 2 * half;          // cols 0..15 (Re)
  const float* b1 = FT + (16 + l16) * 256 + 2 * half;   // cols 16..31 (Im)
  v8f acc0 = {}, acc1 = {};
#pragma unroll 4
  for (int k0 = 0; k0 < 256; k0 += 4) {
    v2f a = ld2_nt(arow + k0);
    acc0 = wmma4(a, ld2(b0 + k0), acc0);
    acc1 = wmma4(a, ld2(b1 + k0), acc1);
  }
  float* yrow = Y + rowbase * 32;
#pragma unroll
  for (int r = 0; r < 8; ++r) {
    int row = r + half * 8;
    yrow[row * 32 + l16] = acc0[r];
    yrow[row * 32 + 16 + l16] = acc1[r];
  }
}

// ---------------------------------------------------------------------------
// K2: forward partial DFT over H, 32 kept rows, complex.
// One wave per (b,i). Z stored as [ky'(32)][kx(16)][b(16)][i(32)].
// Zre = Sum c*Yre + s*Yim ; Zim = Sum c*Yim - s*Yre   (e^{-i th})
// ---------------------------------------------------------------------------
__global__ __launch_bounds__(256) void k2_dft_h(const float* __restrict__ Y,
                                                const float* __restrict__ E2c,
                                                const float* __restrict__ E2s,
                                                float* __restrict__ Zre,
                                                float* __restrict__ Zim) {
  const int bi = blockIdx.x * 8 + (threadIdx.x >> 5);  // 0..511
  const int lane = threadIdx.x & 31;
  const int half = lane >> 4, l16 = lane & 15;
  const float* yb = Y + (long)bi * 256 * 32;
  const float* c0 = E2c + l16 * 256 + 2 * half;
  const float* c1 = E2c + (16 + l16) * 256 + 2 * half;
  const float* s0 = E2s + l16 * 256 + 2 * half;
  const float* s1 = E2s + (16 + l16) * 256 + 2 * half;
  v8f a0 = {}, a1 = {}, a2 = {}, a3 = {}, a4 = {}, a5 = {}, a6 = {}, a7 = {};
#pragma unroll 2
  for (int k0 = 0; k0 < 256; k0 += 4) {
    int kr = k0 + 2 * half;
    v2f bre, bim;
    bre.x = yb[kr * 32 + l16];
    bre.y = yb[(kr + 1) * 32 + l16];
    bim.x = yb[kr * 32 + 16 + l16];
    bim.y = yb[(kr + 1) * 32 + 16 + l16];
    v2f ac0 = ld2(c0 + k0), ac1 = ld2(c1 + k0);
    v2f as0 = ld2(s0 + k0), as1 = ld2(s1 + k0);
    a0 = wmma4(ac0, bre, a0);
    a1 = wmma4(as0, bim, a1);
    a2 = wmma4(ac0, bim, a2);
    a3 = wmma4(as0, bre, a3);
    a4 = wmma4(ac1, bre, a4);
    a5 = wmma4(as1, bim, a5);
    a6 = wmma4(ac1, bim, a6);
    a7 = wmma4(as1, bre, a7);
  }
  const int b = bi >> 5, i = bi & 31;
#pragma unroll
  for (int r = 0; r < 8; ++r) {
    int kyr = r + half * 8;
    long o0 = (((long)kyr * 16 + l16) * 16 + b) * 32 + i;
    long o1 = (((long)(16 + kyr) * 16 + l16) * 16 + b) * 32 + i;
    Zre[o0] = a0[r] + a1[r];
    Zim[o0] = a2[r] - a3[r];
    Zre[o1] = a4[r] + a5[r];
    Zim[o1] = a6[r] - a7[r];
  }
}

// ---------------------------------------------------------------------------
// K3: complex channel mixing per mode. One wave per mode m = ky'*16+kx.
// T[b][o][ky'][kx], Tre = ZrWr - ZiWi ; Tim = ZrWi + ZiWr.
// ---------------------------------------------------------------------------
__global__ __launch_bounds__(256) void k3_mix(
    const float* __restrict__ Zre, const float* __restrict__ Zim,
    const float* __restrict__ w1r, const float* __restrict__ w1i,
    const float* __restrict__ w2r, const float* __restrict__ w2i,
    float* __restrict__ Tre, float* __restrict__ Tim) {
  const int m = blockIdx.x * 8 + (threadIdx.x >> 5);  // 0..511
  const int lane = threadIdx.x & 31;
  const int half = lane >> 4, l16 = lane & 15;
  const int kyp = m >> 4, kx = m & 15;
  const int ky = kyp & 15;
  const float* wr = (kyp < 16) ? w1r : w2r;
  const float* wi = (kyp < 16) ? w1i : w2i;
  const float* zre = Zre + (long)m * 512 + l16 * 32 + 2 * half;
  const float* zim = Zim + (long)m * 512 + l16 * 32 + 2 * half;
  const long wofs = (long)ky * 16 + kx;  // weights: [i][o][ky][kx]
  v8f p0 = {}, q0 = {}, u0 = {}, v0 = {}, p1 = {}, q1 = {}, u1 = {}, v1 = {};
#pragma unroll
  for (int k0 = 0; k0 < 32; k0 += 4) {
    int kr = k0 + 2 * half;
    v2f azr = ld2(zre + k0);
    v2f azi = ld2(zim + k0);
    long ba0 = ((long)kr * 32 + l16) * 256 + wofs;
    long ba0b = ((long)(kr + 1) * 32 + l16) * 256 + wofs;
    long ba1 = ((long)kr * 32 + 16 + l16) * 256 + wofs;
    long ba1b = ((long)(kr + 1) * 32 + 16 + l16) * 256 + wofs;
    v2f br0, bi0, br1, bi1;
    br0.x = wr[ba0];  br0.y = wr[ba0b];
    bi0.x = wi[ba0];  bi0.y = wi[ba0b];
    br1.x = wr[ba1];  br1.y = wr[ba1b];
    bi1.x = wi[ba1];  bi1.y = wi[ba1b];
    p0 = wmma4(azr, br0, p0);
    q0 = wmma4(azi, bi0, q0);
    u0 = wmma4(azr, bi0, u0);
    v0 = wmma4(azi, br0, v0);
    p1 = wmma4(azr, br1, p1);
    q1 = wmma4(azi, bi1, q1);
    u1 = wmma4(azr, bi1, u1);
    v1 = wmma4(azi, br1, v1);
  }
#pragma unroll
  for (int r = 0; r < 8; ++r) {
    int b = r + half * 8;
    long o0 = (((long)b * 32 + l16) * 32 + kyp) * 16 + kx;
    long o1 = (((long)b * 32 + 16 + l16) * 32 + kyp) * 16 + kx;
    Tre[o0] = p0[r] - q0[r];
    Tim[o0] = u0[r] + v0[r];
    Tre[o1] = p1[r] - q1[r];
    Tim[o1] = u1[r] + v1[r];
  }
}

// ---------------------------------------------------------------------------
// K4: inverse DFT over H (e^{+i th}): Ure = Sum c*Tr - s*Ti ; Uim = c*Ti + s*Tr.
// One wave per (b,o, h-tile). U stored [bo][h][32] (Re cols 0..15, Im 16..31).
// ---------------------------------------------------------------------------
__global__ __launch_bounds__(256) void k4_idft_h(const float* __restrict__ Tre,
                                                 const float* __restrict__ Tim,
                                                 const float* __restrict__ E4c,
                                                 const float* __restrict__ E4s,
                                                 float* __restrict__ U) {
  const int wid = blockIdx.x * 8 + (threadIdx.x >> 5);  // 0..8191
  const int lane = threadIdx.x & 31;
  const int half = lane >> 4, l16 = lane & 15;
  const int bo = wid >> 4;
  const int h0 = (wid & 15) * 16;
  const float* tre = Tre + (long)bo * 512;
  const float* tim = Tim + (long)bo * 512;
  const float* ec = E4c + (h0 + l16) * 32 + 2 * half;
  const float* es = E4s + (h0 + l16) * 32 + 2 * half;
  v8f a0 = {}, a1 = {}, a2 = {}, a3 = {};
#pragma unroll
  for (int k0 = 0; k0 < 32; k0 += 4) {
    int kr = k0 + 2 * half;
    v2f ac = ld2(ec + k0);
    v2f as = ld2(es + k0);
    v2f bre, bim;
    bre.x = tre[kr * 16 + l16];
    bre.y = tre[(kr + 1) * 16 + l16];
    bim.x = tim[kr * 16 + l16];
    bim.y = tim[(kr + 1) * 16 + l16];
    a0 = wmma4(ac, bre, a0);
    a1 = wmma4(as, bim, a1);
    a2 = wmma4(ac, bim, a2);
    a3 = wmma4(as, bre, a3);
  }
  float* ub = U + ((long)bo * 256 + h0) * 32;
#pragma unroll
  for (int r = 0; r < 8; ++r) {
    int hh = r + half * 8;
    ub[hh * 32 + l16] = a0[r] - a1[r];       // Ure
    ub[hh * 32 + 16 + l16] = a2[r] + a3[r];  // Uim
  }
}

// ---------------------------------------------------------------------------
// K5: inverse rFFT over W with Hermitian weights folded into GT.
// GEMM [131072,32]x[32,256]; one wave = 16 rows x 64 cols (4 N-tiles).
// out is written with NT hints (write once, never re-read in dispatch).
// ---------------------------------------------------------------------------
__global__ __launch_bounds__(256) void k5_idft_w(const float* __restrict__ U,
                                                 const float* __restrict__ GT,
                                                 float* __restrict__ out) {
  const int wid = blockIdx.x * 8 + (threadIdx.x >> 5);  // 0..32767
  const int lane = threadIdx.x & 31;
  const int half = lane >> 4, l16 = lane & 15;
  const long row0 = (long)(wid >> 2) * 16;
  const int n0 = (wid & 3) * 64;
  const float* arow = U + (row0 + l16) * 32 + 2 * half;
  const float* g0 = GT + (n0 + l16) * 32 + 2 * half;
  v8f acc[4] = {};
#pragma unroll
  for (int k0 = 0; k0 < 32; k0 += 4) {
    v2f a = ld2(arow + k0);
#pragma unroll
    for (int t = 0; t < 4; ++t) {
      acc[t] = wmma4(a, ld2(g0 + t * 16 * 32 + k0), acc[t]);
    }
  }
  float* orow = out + row0 * 256;
#pragma unroll
  for (int r = 0; r < 8; ++r) {
    int rr = r + half * 8;
#pragma unroll
    for (int t = 0; t < 4; ++t) {
      __builtin_nontemporal_store(acc[t][r],
                                  &orow[(long)rr * 256 + n0 + t * 16 + l16]);
    }
  }
}

// ---------------------------------------------------------------------------
extern "C" void kernel_launch(void* const* d_in, const int* in_sizes, int n_in,
                              void* d_out, int out_size, void* d_ws,
                              size_t ws_size, hipStream_t stream) {
  (void)in_sizes; (void)n_in; (void)out_size; (void)ws_size;
  const float* x = (const float*)d_in[0];
  const float* w1r = (const float*)d_in[1];
  const float* w1i = (const float*)d_in[2];
  const float* w2r = (const float*)d_in[3];
  const float* w2i = (const float*)d_in[4];
  float* out = (float*)d_out;

  float* tab = (float*)d_ws;           // tables: 49152 floats (192 KB)
  float* FT = tab;                     // [32][256]
  float* E2c = tab + 8192;             // [32][256]
  float* E2s = tab + 16384;            // [32][256]
  float* E4c = tab + 24576;            // [256][32]
  float* E4s = tab + 32768;            // [256][32]
  float* GT = tab + 40960;             // [256][32]
  float* Y = tab + 49152;              // [512][256][32]  (16.8 MB)
  float* Zre = Y + 4194304;            // [32][16][16][32] (1 MB)
  float* Zim = Zre + 262144;
  float* Tre = Zim + 262144;           // [16][32][32][16] (1 MB)
  float* Tim = Tre + 262144;
  float* U = Tim + 262144;             // [512][256][32]  (16.8 MB)
  // total workspace use: 9,486,336 floats ~= 38 MB

  hipLaunchKernelGGL(k0_tables, dim3(192), dim3(256), 0, stream, tab);
  hipLaunchKernelGGL(k1_dft_w, dim3(1024), dim3(256), 0, stream, x, FT, Y);
  hipLaunchKernelGGL(k2_dft_h, dim3(64), dim3(256), 0, stream, Y, E2c, E2s, Zre,
                     Zim);
  hipLaunchKernelGGL(k3_mix, dim3(64), dim3(256), 0, stream, Zre, Zim, w1r, w1i,
                     w2r, w2i, Tre, Tim);
  hipLaunchKernelGGL(k4_idft_h, dim3(1024), dim3(256), 0, stream, Tre, Tim, E4c,
                     E4s, U);
  hipLaunchKernelGGL(k5_idft_w, dim3(4096), dim3(256), 0, stream, U, GT, out);
}